// GraphFC_22110491640098
// MI455X (gfx1250) — compile-verified
//
#include <hip/hip_runtime.h>

typedef __attribute__((ext_vector_type(2))) float v2f;
typedef __attribute__((ext_vector_type(8))) float v8f;

#define H 128

__device__ __forceinline__ void atomic_add_f32(float* p, float v) {
  __hip_atomic_fetch_add(p, v, __ATOMIC_RELAXED, __HIP_MEMORY_SCOPE_AGENT);
}

// ---------------- precompute folded parameters (tiny, one block) -------------
// params layout (floats):
//  [0*H..): n1w      [1*H..): u2        [2*H..): g2 = n1_b + v2 + l2_b
//  [3*H..): u3       [4*H..): g3 = v3 + l3_b
//  [5*H..): p = n1_w@n2_w     [6*H..): q = n1_b@n2_w + n2_b
//  [7*H..): w1 = n3_w@dec_w[:128]   [8*H..): w2 = n3_w@dec_w[128:]
//  [9*H+0]: u1   [9*H+1]: v1+l1_b   [9*H+2]: c1   [9*H+3]: c2
__global__ void precompute_kernel(const float* __restrict__ em_w, const float* __restrict__ em_b,
    const float* __restrict__ l1_w, const float* __restrict__ l1_b,
    const float* __restrict__ n1_w, const float* __restrict__ n1_b,
    const float* __restrict__ l2_w, const float* __restrict__ l2_b,
    const float* __restrict__ n2_w, const float* __restrict__ n2_b,
    const float* __restrict__ l3_w, const float* __restrict__ l3_b,
    const float* __restrict__ n3_w, const float* __restrict__ n3_b,
    const float* __restrict__ dec_w, float* __restrict__ params) {
  int j = threadIdx.x;
  if (j < H) {
    float u2 = 0.f, v2 = 0.f, u3 = 0.f, v3 = 0.f, p = 0.f, q = 0.f, w1 = 0.f, w2 = 0.f;
    for (int k = 0; k < H; ++k) {
      float ew = em_w[k], eb = em_b[k];
      u2 += ew * l2_w[k * H + j];
      v2 += eb * l2_w[k * H + j];
      u3 += ew * l3_w[k * H + j];
      v3 += eb * l3_w[k * H + j];
      p  += n1_w[k] * n2_w[k * H + j];
      q  += n1_b[k] * n2_w[k * H + j];
      w1 += n3_w[j * H + k] * dec_w[k];
      w2 += n3_w[j * H + k] * dec_w[H + k];
    }
    params[0 * H + j] = n1_w[j];
    params[1 * H + j] = u2;
    params[2 * H + j] = n1_b[j] + v2 + l2_b[j];
    params[3 * H + j] = u3;
    params[4 * H + j] = v3 + l3_b[j];
    params[5 * H + j] = p;
    params[6 * H + j] = q + n2_b[j];
    params[7 * H + j] = w1;
    params[8 * H + j] = w2;
  }
  if (j == 0) {
    float u1 = 0.f, v1 = 0.f, c1 = 0.f, c2 = 0.f;
    for (int k = 0; k < H; ++k) {
      u1 += em_w[k] * l1_w[k];
      v1 += em_b[k] * l1_w[k];
      c1 += n3_b[k] * dec_w[k];
      c2 += n3_b[k] * dec_w[H + k];
    }
    params[9 * H + 0] = u1;
    params[9 * H + 1] = v1 + l1_b[0];
    params[9 * H + 2] = c1;
    params[9 * H + 3] = c2;
  }
}

// ---------------- t[n] = x[n] (with zero tail padding up to Npad) ------------
__global__ void init_t_kernel(const float* __restrict__ x, float* __restrict__ t,
                              int N, int Npad) {
  int n = blockIdx.x * blockDim.x + threadIdx.x;
  if (n < Npad) t[n] = (n < N) ? x[n] : 0.f;
}

// ---------------- layer-1 edge pass (scalar messages) ------------------------
__global__ void edge_scalar_kernel(const int* __restrict__ ei, const float* __restrict__ ea,
                                   const float* __restrict__ x, const float* __restrict__ params,
                                   float* __restrict__ t, int E) {
  int e = blockIdx.x * blockDim.x + threadIdx.x;
  if (e >= E) return;
  float u1 = params[9 * H + 0], v1 = params[9 * H + 1];
  int src = ei[e], dst = ei[E + e];
  float m = fmaxf(x[src] + ea[e] * u1 + v1, 0.f);
  atomic_add_f32(&t[dst], m);
}

// ---------------- layer-2 edge pass: msg2 = relu(t_s*n1w + a*u2 + g2) --------
__global__ void edge_msg2_kernel(const int* __restrict__ ei, const float* __restrict__ ea,
                                 const float* __restrict__ t, const float* __restrict__ params,
                                 float* __restrict__ agg, int E) {
  int gid = blockIdx.x * blockDim.x + threadIdx.x;
  int e = gid >> 5;
  int lane = threadIdx.x & 31;
  if (e >= E) return;
  int src = ei[e], dst = ei[E + e];
  float a = ea[e];
  float ts = t[src];
  float4 w = ((const float4*)(params + 0 * H))[lane];
  float4 u = ((const float4*)(params + 1 * H))[lane];
  float4 g = ((const float4*)(params + 2 * H))[lane];
  float m0 = fmaxf(ts * w.x + a * u.x + g.x, 0.f);
  float m1 = fmaxf(ts * w.y + a * u.y + g.y, 0.f);
  float m2 = fmaxf(ts * w.z + a * u.z + g.z, 0.f);
  float m3 = fmaxf(ts * w.w + a * u.w + g.w, 0.f);
  float* d = agg + (size_t)dst * H + lane * 4;
  atomic_add_f32(d + 0, m0);
  atomic_add_f32(d + 1, m1);
  atomic_add_f32(d + 2, m2);
  atomic_add_f32(d + 3, m3);
}

// ---------------- hdn2 = t*p + q + agg2 @ n2_w  (fp32 WMMA GEMM) -------------
// One wave per 16x16 output tile; K=128 via 32 x v_wmma_f32_16x16x4_f32.
__global__ void node_mm_wmma_kernel(const float* __restrict__ t, const float* __restrict__ agg2,
                                    const float* __restrict__ n2w, const float* __restrict__ params,
                                    float* __restrict__ hdn2, int nTilesM) {
  int waveId = threadIdx.x >> 5;
  int lane = threadIdx.x & 31;
  int tile = blockIdx.x * (blockDim.x >> 5) + waveId;
  int totalTiles = nTilesM * (H / 16);
  if (tile >= totalTiles) return;           // wave-uniform; EXEC stays all-ones
  int tn = tile & (H / 16 - 1);
  int tm = tile >> 3;                       // H/16 == 8
  int lm = lane & 15;
  int half = lane >> 4;
  int n = tn * 16 + lm;
  float pn = params[5 * H + n];
  float qn = params[6 * H + n];
  int mbase = tm * 16;
  v8f acc;
#pragma unroll
  for (int v = 0; v < 8; ++v) {
    int m = mbase + v + half * 8;           // C/D layout: VGPR v, half picks M or M+8
    acc[v] = t[m] * pn + qn;
  }
  const float* arow = agg2 + (size_t)(mbase + lm) * H;
#pragma unroll
  for (int k0 = 0; k0 < H; k0 += 4) {
    int ka = k0 + half * 2;                 // lanes 16-31 carry the upper K half
    v2f a, b;
    a.x = arow[ka];
    a.y = arow[ka + 1];
    b.x = n2w[(size_t)ka * H + n];
    b.y = n2w[(size_t)(ka + 1) * H + n];
    acc = __builtin_amdgcn_wmma_f32_16x16x4_f32(false, a, false, b, (short)0, acc,
                                                false, false);
  }
#pragma unroll
  for (int v = 0; v < 8; ++v) {
    int m = mbase + v + half * 8;
    hdn2[(size_t)m * H + n] = acc[v];
  }
}

// ---------------- layer-3 edge pass: msg3 = relu(hdn2[src] + a*u3 + g3) ------
__global__ void edge_msg3_kernel(const int* __restrict__ ei, const float* __restrict__ ea,
                                 const float* __restrict__ hdn2, const float* __restrict__ params,
                                 float* __restrict__ agg, int E) {
  int gid = blockIdx.x * blockDim.x + threadIdx.x;
  int e = gid >> 5;
  int lane = threadIdx.x & 31;
  if (e >= E) return;
  int src = ei[e], dst = ei[E + e];
  float a = ea[e];
  float4 hv = *(const float4*)(hdn2 + (size_t)src * H + lane * 4);
  float4 u = ((const float4*)(params + 3 * H))[lane];
  float4 g = ((const float4*)(params + 4 * H))[lane];
  float m0 = fmaxf(hv.x + a * u.x + g.x, 0.f);
  float m1 = fmaxf(hv.y + a * u.y + g.y, 0.f);
  float m2 = fmaxf(hv.z + a * u.z + g.z, 0.f);
  float m3 = fmaxf(hv.w + a * u.w + g.w, 0.f);
  float* d = agg + (size_t)dst * H + lane * 4;
  atomic_add_f32(d + 0, m0);
  atomic_add_f32(d + 1, m1);
  atomic_add_f32(d + 2, m2);
  atomic_add_f32(d + 3, m3);
}

// ---------------- per-node folded decoder: s1/s2 dots ------------------------
__global__ void node_dec_kernel(const float* __restrict__ hdn2, const float* __restrict__ agg3,
                                const float* __restrict__ params, float* __restrict__ s1,
                                float* __restrict__ s2, int N) {
  int n = blockIdx.x * blockDim.x + threadIdx.x;
  if (n >= N) return;
  float acc1 = params[9 * H + 2];
  float acc2 = params[9 * H + 3];
  const float4* hv = (const float4*)(hdn2 + (size_t)n * H);
  const float4* av = (const float4*)(agg3 + (size_t)n * H);
  const float4* w1 = (const float4*)(params + 7 * H);
  const float4* w2 = (const float4*)(params + 8 * H);
#pragma unroll
  for (int j = 0; j < H / 4; ++j) {
    float4 h = hv[j], a = av[j], x1 = w1[j], x2 = w2[j];
    float sx = h.x + a.x, sy = h.y + a.y, sz = h.z + a.z, sw = h.w + a.w;
    acc1 += sx * x1.x + sy * x1.y + sz * x1.z + sw * x1.w;
    acc2 += sx * x2.x + sy * x2.y + sz * x2.z + sw * x2.w;
  }
  s1[n] = acc1;
  s2[n] = acc2;
}

// ---------------- final per-edge output --------------------------------------
__global__ void edge_out_kernel(const int* __restrict__ ei, const float* __restrict__ s1,
                                const float* __restrict__ s2, const float* __restrict__ dec_b,
                                float* __restrict__ out, int E) {
  int e = blockIdx.x * blockDim.x + threadIdx.x;
  if (e >= E) return;
  out[e] = s1[ei[e]] + s2[ei[E + e]] + dec_b[0];
}

extern "C" void kernel_launch(void* const* d_in, const int* in_sizes, int n_in,
                              void* d_out, int out_size, void* d_ws, size_t ws_size,
                              hipStream_t stream) {
  const float* x    = (const float*)d_in[0];
  const int*   ei   = (const int*)d_in[1];
  const float* ea   = (const float*)d_in[2];
  const float* em_w = (const float*)d_in[3];
  const float* em_b = (const float*)d_in[4];
  const float* l1_w = (const float*)d_in[5];
  const float* l1_b = (const float*)d_in[6];
  const float* n1_w = (const float*)d_in[7];
  const float* n1_b = (const float*)d_in[8];
  const float* l2_w = (const float*)d_in[9];
  const float* l2_b = (const float*)d_in[10];
  const float* n2_w = (const float*)d_in[11];
  const float* n2_b = (const float*)d_in[12];
  const float* l3_w = (const float*)d_in[13];
  const float* l3_b = (const float*)d_in[14];
  const float* n3_w = (const float*)d_in[15];
  const float* n3_b = (const float*)d_in[16];
  const float* dec_w = (const float*)d_in[17];
  const float* dec_b = (const float*)d_in[18];
  float* out = (float*)d_out;

  int N = in_sizes[0];
  int E = in_sizes[2];
  int Npad = ((N + 15) / 16) * 16;

  float* ws = (float*)d_ws;
  float* params = ws;                               // 2048 floats reserved
  float* t    = ws + 2048;                          // Npad
  float* s1   = t + Npad;                           // Npad
  float* s2   = s1 + Npad;                          // Npad
  float* bufA = s2 + Npad;                          // Npad*H (agg2 then agg3)
  float* hdn2 = bufA + (size_t)Npad * H;            // Npad*H

  precompute_kernel<<<1, 128, 0, stream>>>(em_w, em_b, l1_w, l1_b, n1_w, n1_b,
                                           l2_w, l2_b, n2_w, n2_b, l3_w, l3_b,
                                           n3_w, n3_b, dec_w, params);

  init_t_kernel<<<(Npad + 255) / 256, 256, 0, stream>>>(x, t, N, Npad);
  edge_scalar_kernel<<<(E + 255) / 256, 256, 0, stream>>>(ei, ea, x, params, t, E);

  hipMemsetAsync(bufA, 0, (size_t)Npad * H * sizeof(float), stream);
  {
    long long threads = (long long)E * 32;
    edge_msg2_kernel<<<(unsigned)((threads + 255) / 256), 256, 0, stream>>>(
        ei, ea, t, params, bufA, E);
  }

  int nTilesM = Npad / 16;
  int totalTiles = nTilesM * (H / 16);
  node_mm_wmma_kernel<<<(totalTiles + 7) / 8, 256, 0, stream>>>(t, bufA, n2_w, params,
                                                                hdn2, nTilesM);

  hipMemsetAsync(bufA, 0, (size_t)Npad * H * sizeof(float), stream);
  {
    long long threads = (long long)E * 32;
    edge_msg3_kernel<<<(unsigned)((threads + 255) / 256), 256, 0, stream>>>(
        ei, ea, hdn2, params, bufA, E);
  }

  node_dec_kernel<<<(N + 255) / 256, 256, 0, stream>>>(hdn2, bufA, params, s1, s2, N);
  edge_out_kernel<<<(E + 255) / 256, 256, 0, stream>>>(ei, s1, s2, dec_b, out, E);
}